// GRNN_25881472926277
// MI455X (gfx1250) — compile-verified
//
#include <hip/hip_runtime.h>
#include <hip/hip_bf16.h>

// ---------------------------------------------------------------------------
// GCN (2-layer) for MI455X / gfx1250.
//  h   = X @ W                      (WMMA f32 16x16x4, W staged in LDS,
//                                    K-paired so B frags are single b64 loads)
//  agg[dst] += h[src]*dis[src]*dis[dst]   (per-edge wave, f32 atomics in L2)
//  out = relu(agg + h*dis^2 + b)    (self-loop folded into epilogue)
// ---------------------------------------------------------------------------

typedef __attribute__((ext_vector_type(2))) float v2f;
typedef __attribute__((ext_vector_type(8))) float v8f;

#define DIM 128  // IN_DIM == HID_DIM == 128

// ---------------- utility: zero a buffer (float4 granularity) --------------
__global__ void gcn_zero4(float* __restrict__ p, int n4) {
  int i = blockIdx.x * blockDim.x + threadIdx.x;
  if (i < n4) ((float4*)p)[i] = make_float4(0.f, 0.f, 0.f, 0.f);
}

// ---------------- degree count (self-loop added later as +1) ---------------
__global__ void gcn_deg(const int* __restrict__ dst, int* __restrict__ deg,
                        int nEdges) {
  int e = blockIdx.x * blockDim.x + threadIdx.x;
  if (e < nEdges) atomicAdd(&deg[dst[e]], 1);
}

__global__ void gcn_dis(const int* __restrict__ deg, float* __restrict__ dis,
                        int n) {
  int i = blockIdx.x * blockDim.x + threadIdx.x;
  if (i < n) dis[i] = rsqrtf((float)(deg[i] + 1));  // +1 = self loop
}

// ---------------- WMMA GEMM: H = X @ W  ([n x 128] @ [128 x 128]) ----------
// Block: 256 threads = 8 waves; block covers 128 rows; wave w -> rows
// [base + 16w, +16), all 128 output columns (8 accumulator tiles).
// LDS holds W in K-paired layout: Wp[p][n] = {W[2p][n], W[2p+1][n]} so each
// WMMA B fragment is one aligned ds_load_b64 (no register shuffles).
__global__ __launch_bounds__(256) void gcn_gemm_wmma(
    const float* __restrict__ X, const float* __restrict__ W,
    float* __restrict__ H, int nRows) {
  __shared__ float2 Wp[(DIM / 2) * DIM];  // 64 KB of the 320 KB WGP LDS

  const int tid = threadIdx.x;
  {  // cooperative load: coalesced float4 reads, paired ds_store_b128 writes
    const float4* Wv = (const float4*)W;
#pragma unroll
    for (int i = 0; i < 8; ++i) {
      const int id = tid + i * 256;  // 0..2047 (p, n4) work items
      const int p = id >> 5;         // K-pair index 0..63
      const int n4 = id & 31;        // float4 column group
      const float4 a = Wv[(2 * p) * (DIM / 4) + n4];      // row 2p
      const float4 c = Wv[(2 * p + 1) * (DIM / 4) + n4];  // row 2p+1
      float4* o = (float4*)&Wp[p * DIM + n4 * 4];
      o[0] = make_float4(a.x, c.x, a.y, c.y);
      o[1] = make_float4(a.z, c.z, a.w, c.w);
    }
  }
  __syncthreads();

  const int wave = tid >> 5;
  const int lane = tid & 31;
  const int ll = lane & 15;   // low-lane index (M for A, N for B/C/D)
  const int hi = lane >> 4;   // half-wave select (K pair for A/B, M+8 for C/D)
  const int rowBase = blockIdx.x * 128 + wave * 16;

  v8f acc[8];
#pragma unroll
  for (int nt = 0; nt < 8; ++nt) {
    v8f z = {};
    acc[nt] = z;
  }

  // A-fragment row (clamped so tail blocks keep EXEC all-ones for WMMA)
  int arow = rowBase + ll;
  if (arow > nRows - 1) arow = nRows - 1;
  const float* Xrow = X + (size_t)arow * DIM;

#pragma unroll 4
  for (int ks = 0; ks < 32; ++ks) {  // K = 128 in steps of 4
    const int k0 = ks * 4;
    // A 16x4 f32: lanes 0-15 hold K0,K1; lanes 16-31 hold K2,K3
    v2f a = *(const v2f*)(Xrow + k0 + 2 * hi);
    // B 4x16: lane = N; this lane's K pair is row (2*ks + hi) of Wp
    const float2* brow = Wp + (2 * ks + hi) * DIM;
#pragma unroll
    for (int nt = 0; nt < 8; ++nt) {
      v2f b = *(const v2f*)(brow + nt * 16 + ll);  // single ds_load_b64
      acc[nt] = __builtin_amdgcn_wmma_f32_16x16x4_f32(
          false, a, false, b, (short)0, acc[nt], false, false);
    }
  }

  // C/D layout: VGPR j -> M = j (lanes 0-15) or M = j+8 (lanes 16-31); N = ll
#pragma unroll
  for (int j = 0; j < 8; ++j) {
    const int row = rowBase + hi * 8 + j;
    if (row < nRows) {
      float* o = H + (size_t)row * DIM + ll;
#pragma unroll
      for (int nt = 0; nt < 8; ++nt) o[nt * 16] = acc[nt][j];
    }
  }
}

// ---------------- edge aggregation: AGG[dst] += H[src] * norm --------------
// One wave per edge; lane handles 4 consecutive features (float4 gather,
// 4x global_atomic_add_f32 — dst rows live in the 192MB L2).
__global__ __launch_bounds__(256) void gcn_scatter(
    const float* __restrict__ H, const int* __restrict__ src,
    const int* __restrict__ dst, const float* __restrict__ dis,
    float* __restrict__ AGG, int nEdges) {
  const int e = blockIdx.x * (blockDim.x >> 5) + (threadIdx.x >> 5);
  if (e >= nEdges) return;  // wave-uniform exit
  const int lane = threadIdx.x & 31;
  const int s = __builtin_amdgcn_readfirstlane(src[e]);
  const int d = __builtin_amdgcn_readfirstlane(dst[e]);
  const float norm = dis[s] * dis[d];
  const float4 h4 = *(const float4*)(H + (size_t)s * DIM + lane * 4);
  float* o = AGG + (size_t)d * DIM + lane * 4;
  atomicAdd(o + 0, h4.x * norm);
  atomicAdd(o + 1, h4.y * norm);
  atomicAdd(o + 2, h4.z * norm);
  atomicAdd(o + 3, h4.w * norm);
}

// ---------------- epilogue: OUT = relu(AGG + H*dis^2 + b) ------------------
__global__ void gcn_finalize(const float* __restrict__ AGG,
                             const float* __restrict__ H,
                             const float* __restrict__ dis,
                             const float* __restrict__ b,
                             float* __restrict__ OUT, int nNodes) {
  int t = blockIdx.x * blockDim.x + threadIdx.x;  // one float4 chunk each
  if (t >= nNodes * (DIM / 4)) return;
  const int node = t >> 5;
  const int c = (t & 31) * 4;
  float dd = dis[node];
  dd = dd * dd;  // self-loop norm = dis[i]^2
  const float4 a4 = *(const float4*)(AGG + (size_t)node * DIM + c);
  const float4 h4 = *(const float4*)(H + (size_t)node * DIM + c);
  const float4 b4 = *(const float4*)(b + c);
  float4 o;
  o.x = fmaxf(fmaf(h4.x, dd, a4.x) + b4.x, 0.f);
  o.y = fmaxf(fmaf(h4.y, dd, a4.y) + b4.y, 0.f);
  o.z = fmaxf(fmaf(h4.z, dd, a4.z) + b4.z, 0.f);
  o.w = fmaxf(fmaf(h4.w, dd, a4.w) + b4.w, 0.f);
  *(float4*)(OUT + (size_t)node * DIM + c) = o;
}

// ---------------------------------------------------------------------------
extern "C" void kernel_launch(void* const* d_in, const int* in_sizes, int n_in,
                              void* d_out, int out_size, void* d_ws,
                              size_t ws_size, hipStream_t stream) {
  const float* x = (const float*)d_in[0];
  const int* eidx = (const int*)d_in[1];
  const float* W1 = (const float*)d_in[2];
  const float* b1 = (const float*)d_in[3];
  const float* W2 = (const float*)d_in[4];
  const float* b2 = (const float*)d_in[5];
  float* out = (float*)d_out;

  const int nNodes = in_sizes[0] / DIM;
  const int nEdges = in_sizes[1] / 2;
  const int* src = eidx;           // edge_index[0]
  const int* dst = eidx + nEdges;  // edge_index[1]

  // workspace carve-up (256B aligned)
  char* ws = (char*)d_ws;
  auto align256 = [](size_t v) { return (v + 255) & ~(size_t)255; };
  int* deg = (int*)ws;
  ws += align256((size_t)nNodes * sizeof(int));
  float* dis = (float*)ws;
  ws += align256((size_t)nNodes * sizeof(float));
  float* bufA = (float*)ws;  // h (GEMM output)
  ws += align256((size_t)nNodes * DIM * sizeof(float));
  float* bufB = (float*)ws;  // layer-1 aggregation / activations

  const int feat4 = nNodes * (DIM / 4);  // float4 chunks in a feature matrix
  const int zThreads = 256;

  // --- init: zero deg, layer-1 agg buffer, and d_out (layer-2 agg) ---------
  gcn_zero4<<<(nNodes / 4 + zThreads - 1) / zThreads, zThreads, 0, stream>>>(
      (float*)deg, nNodes / 4);
  gcn_zero4<<<(feat4 + zThreads - 1) / zThreads, zThreads, 0, stream>>>(bufB,
                                                                        feat4);
  gcn_zero4<<<(feat4 + zThreads - 1) / zThreads, zThreads, 0, stream>>>(out,
                                                                        feat4);

  // --- normalization coefficients ------------------------------------------
  gcn_deg<<<(nEdges + 255) / 256, 256, 0, stream>>>(dst, deg, nEdges);
  gcn_dis<<<(nNodes + 255) / 256, 256, 0, stream>>>(deg, dis, nNodes);

  const int gemmBlocks = (nNodes + 127) / 128;
  const int scatBlocks = (nEdges + 7) / 8;
  const int finBlocks = (feat4 + 255) / 256;

  // --- layer 1 --------------------------------------------------------------
  gcn_gemm_wmma<<<gemmBlocks, 256, 0, stream>>>(x, W1, bufA, nNodes);
  gcn_scatter<<<scatBlocks, 256, 0, stream>>>(bufA, src, dst, dis, bufB,
                                              nEdges);
  gcn_finalize<<<finBlocks, 256, 0, stream>>>(bufB, bufA, dis, b1, bufB,
                                              nNodes);

  // --- layer 2 (aggregation goes straight into d_out) ----------------------
  gcn_gemm_wmma<<<gemmBlocks, 256, 0, stream>>>(bufB, W2, bufA, nNodes);
  gcn_scatter<<<scatBlocks, 256, 0, stream>>>(bufA, src, dst, dis, out,
                                              nEdges);
  gcn_finalize<<<finBlocks, 256, 0, stream>>>(out, bufA, dis, b2, out, nNodes);
}